// VGAE_4561255268671
// MI455X (gfx1250) — compile-verified
//
#include <hip/hip_runtime.h>

typedef __attribute__((ext_vector_type(2))) float v2f;
typedef __attribute__((ext_vector_type(8))) float v8f;

#define N_ENT 20000
#define N_REL 400
#define N_TIME 366
#define N_META 800
#define NE 200000
#define NPE 80000

// ---------------------------------------------------------------------------
// One K-chunk (4 k values) of a 16x160 WMMA tile: 1 gathered A float2 per
// lane, then ALL 10 packed-W B fragments loaded into distinct registers
// (clause-able, latency overlapped), then 10 v_wmma_f32_16x16x4_f32.
// ---------------------------------------------------------------------------
__device__ __forceinline__ void kstep(const float* __restrict__ ap, int cb,
                                      int hi, int m, float msk,
                                      const float2* __restrict__ wchunk,
                                      v8f acc[10])
{
    float2 av = *(const float2*)(ap + cb + hi * 2);
    float2 bv[10];
#pragma unroll
    for (int nt = 0; nt < 10; ++nt)
        bv[nt] = wchunk[(nt * 16 + m) * 2 + hi];
    v2f a; a[0] = av.x * msk; a[1] = av.y * msk;
#pragma unroll
    for (int nt = 0; nt < 10; ++nt) {
        v2f b; b[0] = bv[nt].x; b[1] = bv[nt].y;
        acc[nt] = __builtin_amdgcn_wmma_f32_16x16x4_f32(
            false, a, false, b, (short)0, acc[nt], false, false);
    }
}

// ---------------------------------------------------------------------------
// Edge message kernel: one wave per 16-edge tile, full N=160 output.
// comp = [s0[idx0] (w0) | s1[idx1] (w1) | s2[idx2] (w2)], K = w0+w1+w2.
// msg = comp @ W{I|O} + b{I|O} selected per edge class (partitioned by perm),
// scattered with atomic f32 adds into sumBuf[dst].
// ---------------------------------------------------------------------------
__global__ __launch_bounds__(32) void edge_msg_kernel(
    const float* __restrict__ s0, int w0,
    const float* __restrict__ s1, int w1,
    const float* __restrict__ s2, int w2,
    const int* __restrict__ idx0,
    const int* __restrict__ idx1,
    const int* __restrict__ idx2,
    const int* __restrict__ dstArr,
    const int* __restrict__ perm,
    const int* __restrict__ nIptr,
    const float* __restrict__ WpI,   // packed [K/4][160][4]
    const float* __restrict__ WpO,
    const float* __restrict__ bI,
    const float* __restrict__ bO,
    float* __restrict__ sumBuf)
{
    const int N = 160, NT = 10;
    int lane = threadIdx.x;
    int m  = lane & 15;
    int hi = lane >> 4;
    int tileBase = blockIdx.x * 16;
    int nI = *nIptr;

    __shared__ int s_dst[16];
    __shared__ int s_cls[16];

    int pos = tileBase + m;
    int e = perm[pos];
    const float* a0 = s0 + (size_t)idx0[e] * w0;
    const float* a1 = s1 + (size_t)idx1[e] * w1;
    const float* a2 = (w2 > 0) ? (s2 + (size_t)idx2[e] * w2) : s1;
    int cls = (pos >= nI) ? 1 : 0;
    if (lane < 16) { s_dst[m] = dstArr[e]; s_cls[m] = cls; }
    __syncthreads();

    v8f acc[NT];
#pragma unroll
    for (int nt = 0; nt < NT; ++nt)
#pragma unroll
        for (int q = 0; q < 8; ++q) acc[nt][q] = 0.0f;

    int hasI = (tileBase < nI);
    int hasO = (tileBase + 15 >= nI);

    for (int sel = 0; sel < 2; ++sel) {
        if (sel == 0 && !hasI) continue;
        if (sel == 1 && !hasO) continue;
        const float2* wchunk = (const float2*)(sel ? WpO : WpI);
        float msk = (cls == sel) ? 1.0f : 0.0f;
        // three contiguous concat regions, fixed source pointer each: no
        // per-iteration region-select branches in the WMMA loop.
        for (int kb = 0; kb < w0; kb += 4) { kstep(a0, kb, hi, m, msk, wchunk, acc); wchunk += 320; }
        for (int kb = 0; kb < w1; kb += 4) { kstep(a1, kb, hi, m, msk, wchunk, acc); wchunk += 320; }
        for (int kb = 0; kb < w2; kb += 4) { kstep(a2, kb, hi, m, msk, wchunk, acc); wchunk += 320; }
    }

#pragma unroll
    for (int nt = 0; nt < NT; ++nt) {
        int col = nt * 16 + m;
        float biasI = bI[col], biasO = bO[col];
#pragma unroll
        for (int j = 0; j < 8; ++j) {
            int row = j + hi * 8;
            float v = acc[nt][j] + (s_cls[row] ? biasO : biasI);
            atomicAdd(&sumBuf[(size_t)s_dst[row] * N + col], v);
        }
    }
}

// ---------------------------------------------------------------------------
// Dense kernel: C[M,N] = act(A[M,K] @ Wp + bias + sumBuf*recip)
// ---------------------------------------------------------------------------
template<int N>
__global__ __launch_bounds__(32) void dense_kernel(
    const float* __restrict__ A,
    const float* __restrict__ Wp,     // packed [K/4][N][4]
    const float* __restrict__ bias,
    const float* __restrict__ sumBuf, // nullable
    const float* __restrict__ recip,  // nullable
    float* __restrict__ C,
    int M, int K, int relu)
{
    const int NT = N / 16;
    int lane = threadIdx.x;
    int m  = lane & 15;
    int hi = lane >> 4;
    int tileBase = blockIdx.x * 16;
    int arow = tileBase + m;
    if (arow > M - 1) arow = M - 1;
    const float* ap = A + (size_t)arow * K;

    v8f acc[NT];
#pragma unroll
    for (int nt = 0; nt < NT; ++nt)
#pragma unroll
        for (int q = 0; q < 8; ++q) acc[nt][q] = 0.0f;

    const float2* wchunk = (const float2*)Wp;
    for (int kb = 0; kb < K; kb += 4) {
        float2 av = *(const float2*)(ap + kb + hi * 2);
        float2 bv[NT];
#pragma unroll
        for (int nt = 0; nt < NT; ++nt)
            bv[nt] = wchunk[(nt * 16 + m) * 2 + hi];
        v2f a; a[0] = av.x; a[1] = av.y;
#pragma unroll
        for (int nt = 0; nt < NT; ++nt) {
            v2f b; b[0] = bv[nt].x; b[1] = bv[nt].y;
            acc[nt] = __builtin_amdgcn_wmma_f32_16x16x4_f32(
                false, a, false, b, (short)0, acc[nt], false, false);
        }
        wchunk += 2 * N;   // next [N][4] chunk, in float2 units
    }

#pragma unroll
    for (int nt = 0; nt < NT; ++nt) {
        int col = nt * 16 + m;
        float bb = bias[col];
#pragma unroll
        for (int j = 0; j < 8; ++j) {
            int row = tileBase + j + hi * 8;
            if (row < M) {
                float v = acc[nt][j] + bb;
                if (sumBuf) v += sumBuf[(size_t)row * N + col] * recip[row];
                if (relu) v = v > 0.0f ? v : 0.0f;
                C[(size_t)row * N + col] = v;
            }
        }
    }
}

// ---------------------------------------------------------------------------
// Weight repack: W[K][N] -> Wp[k/4][n] = float4{W[4k..4k+3][n]}
// ---------------------------------------------------------------------------
__global__ void repack_kernel(const float* __restrict__ W, float* __restrict__ Wp,
                              int Ktot, int N)
{
    int t = blockIdx.x * blockDim.x + threadIdx.x;
    int total = (Ktot >> 2) * N;
    if (t >= total) return;
    int kc = t / N, n = t - kc * N;
    const float* w = W + (size_t)(kc * 4) * N + n;
    float4 v = make_float4(w[0], w[N], w[2 * N], w[3 * N]);
    ((float4*)Wp)[t] = v;
}

// ---------------------------------------------------------------------------
// In-degree counts and reciprocal (mean-agg with zero for isolated nodes)
// ---------------------------------------------------------------------------
__global__ void count_dst_kernel(const int* __restrict__ d, int n, int* __restrict__ cnt)
{
    int i = blockIdx.x * blockDim.x + threadIdx.x;
    if (i < n) atomicAdd(&cnt[d[i]], 1);
}
__global__ void recip_kernel(const int* __restrict__ cnt, float* __restrict__ r, int n)
{
    int i = blockIdx.x * blockDim.x + threadIdx.x;
    if (i < n) r[i] = cnt[i] > 0 ? 1.0f / (float)cnt[i] : 0.0f;
}

// ---------------------------------------------------------------------------
// Deterministic stable partition by inv (zeros first): block scan machinery
// ---------------------------------------------------------------------------
__global__ void scan_block_count(const int* __restrict__ inv, int n, int* __restrict__ blkCnt)
{
    __shared__ int sh[256];
    int tid = threadIdx.x;
    int i = blockIdx.x * 256 + tid;
    int z = (i < n && inv[i] == 0) ? 1 : 0;
    sh[tid] = z;
    __syncthreads();
    for (int s = 128; s > 0; s >>= 1) {
        if (tid < s) sh[tid] += sh[tid + s];
        __syncthreads();
    }
    if (tid == 0) blkCnt[blockIdx.x] = sh[0];
}
__global__ void scan_exclusive_single(int* __restrict__ blkCnt, int nb, int* __restrict__ totalOut)
{
    if (threadIdx.x == 0 && blockIdx.x == 0) {
        int run = 0;
        for (int i = 0; i < nb; ++i) { int c = blkCnt[i]; blkCnt[i] = run; run += c; }
        *totalOut = run;
    }
}
__global__ void scatter_partition(const int* __restrict__ inv, int n,
                                  const int* __restrict__ blkOff,
                                  const int* __restrict__ total,
                                  int* __restrict__ perm)
{
    __shared__ int sh[256];
    int tid = threadIdx.x;
    int i = blockIdx.x * 256 + tid;
    int z = (i < n && inv[i] == 0) ? 1 : 0;
    sh[tid] = z;
    __syncthreads();
    for (int off = 1; off < 256; off <<= 1) {
        int v = (tid >= off) ? sh[tid - off] : 0;
        __syncthreads();
        sh[tid] += v;
        __syncthreads();
    }
    int excl = sh[tid] - z;
    if (i < n) {
        int zerosBefore = blkOff[blockIdx.x] + excl;
        int pos = z ? zerosBefore : (*total + (i - zerosBefore));
        perm[pos] = i;
    }
}

// ---------------------------------------------------------------------------
// Reparameterization: mean += noise * exp(log_std), in place
// ---------------------------------------------------------------------------
__global__ void sample_kernel(float* __restrict__ mean, const float* __restrict__ logstd,
                              const float* __restrict__ noise, int n)
{
    int i = blockIdx.x * blockDim.x + threadIdx.x;
    if (i < n) mean[i] = mean[i] + noise[i] * expf(logstd[i]);
}

// ---------------------------------------------------------------------------
extern "C" void kernel_launch(void* const* d_in, const int* in_sizes, int n_in,
                              void* d_out, int out_size, void* d_ws, size_t ws_size,
                              hipStream_t stream)
{
    (void)in_sizes; (void)n_in; (void)out_size; (void)ws_size;

    const float* ent_feat = (const float*)d_in[0];
    const float* rel_feat = (const float*)d_in[1];
    const float* time_emb = (const float*)d_in[2];
    const float* meta_emb = (const float*)d_in[3];
    const float* noise_e  = (const float*)d_in[4];
    const float* noise_r  = (const float*)d_in[5];
    const float* entWIw = (const float*)d_in[6];
    const float* entWIb = (const float*)d_in[7];
    const float* entWOw = (const float*)d_in[8];
    const float* entWOb = (const float*)d_in[9];
    const float* entWSw = (const float*)d_in[10];
    const float* entWSb = (const float*)d_in[11];
    const float* entWTw = (const float*)d_in[12];
    const float* entWTb = (const float*)d_in[13];
    const float* relWIw = (const float*)d_in[14];
    const float* relWIb = (const float*)d_in[15];
    const float* relWOw = (const float*)d_in[16];
    const float* relWOb = (const float*)d_in[17];
    const float* relWSw = (const float*)d_in[18];
    const float* relWSb = (const float*)d_in[19];
    const float* relWMw = (const float*)d_in[20];
    const float* relWMb = (const float*)d_in[21];
    const int* srcA  = (const int*)d_in[22];
    const int* dstA  = (const int*)d_in[23];
    const int* eRel  = (const int*)d_in[24];
    const int* eTime = (const int*)d_in[25];
    const int* eInv  = (const int*)d_in[26];
    const int* pSrc  = (const int*)d_in[27];
    const int* pDst  = (const int*)d_in[28];
    const int* pRel  = (const int*)d_in[29];
    const int* pInv  = (const int*)d_in[30];

    float* dout = (float*)d_out;
    float* outEnt   = dout;
    float* outEntO  = dout + 3200000;
    float* outRel   = dout + 6400000;
    float* outRelO  = dout + 6464000;
    float* outTim   = dout + 6528000;
    float* outTimO  = dout + 6557280;
    float* outMeta  = dout + 6586560;
    float* outMetaO = dout + 6714560;

    // ---- workspace carving ----
    size_t off = 0;
    char* base = (char*)d_ws;
    auto take = [&](size_t nbytes) -> void* {
        off = (off + 255) & ~(size_t)255;
        void* p = base + off;
        off += nbytes;
        return p;
    };
    float* sum_e  = (float*)take((size_t)N_ENT * 160 * 4);
    float* sum_p  = (float*)take((size_t)N_REL * 160 * 4);
    float* bufE_A = (float*)take((size_t)N_ENT * 160 * 4);
    float* bufE_B = (float*)take((size_t)N_ENT * 160 * 4);
    float* bufR_A = (float*)take((size_t)N_REL * 160 * 4);
    float* bufR_B = (float*)take((size_t)N_REL * 160 * 4);
    float* bufT   = (float*)take((size_t)N_TIME * 80 * 4);
    float* bufM   = (float*)take((size_t)N_META * 160 * 4);
    int*   cnt_e  = (int*)take(N_ENT * 4);
    int*   cnt_p  = (int*)take(N_REL * 4);
    float* recip_e = (float*)take(N_ENT * 4);
    float* recip_p = (float*)take(N_REL * 4);
    int*   perm_e = (int*)take(NE * 4);
    int*   perm_p = (int*)take(NPE * 4);
    int nbE = (NE + 255) / 256, nbP = (NPE + 255) / 256;
    int* blk_e = (int*)take(nbE * 4);
    int* blk_p = (int*)take(nbP * 4);
    int* nI_e = (int*)take(4);
    int* nI_p = (int*)take(4);
    float* entWIp = (float*)take((size_t)6 * 400 * 160 * 4);
    float* entWOp = (float*)take((size_t)6 * 400 * 160 * 4);
    float* relWIp = (float*)take((size_t)6 * 320 * 160 * 4);
    float* relWOp = (float*)take((size_t)6 * 320 * 160 * 4);
    float* entWSp = (float*)take((size_t)6 * 160 * 160 * 4);
    float* entWTp = (float*)take((size_t)6 * 80 * 80 * 4);
    float* relWSp = (float*)take((size_t)6 * 160 * 160 * 4);
    float* relWMp = (float*)take((size_t)6 * 160 * 160 * 4);

    // ---- setup: counts, reciprocals, partitions, weight repack ----
    hipMemsetAsync(cnt_e, 0, N_ENT * 4, stream);
    hipMemsetAsync(cnt_p, 0, N_REL * 4, stream);
    count_dst_kernel<<<nbE, 256, 0, stream>>>(dstA, NE, cnt_e);
    count_dst_kernel<<<nbP, 256, 0, stream>>>(pDst, NPE, cnt_p);
    recip_kernel<<<(N_ENT + 255) / 256, 256, 0, stream>>>(cnt_e, recip_e, N_ENT);
    recip_kernel<<<(N_REL + 255) / 256, 256, 0, stream>>>(cnt_p, recip_p, N_REL);

    scan_block_count<<<nbE, 256, 0, stream>>>(eInv, NE, blk_e);
    scan_exclusive_single<<<1, 1, 0, stream>>>(blk_e, nbE, nI_e);
    scatter_partition<<<nbE, 256, 0, stream>>>(eInv, NE, blk_e, nI_e, perm_e);
    scan_block_count<<<nbP, 256, 0, stream>>>(pInv, NPE, blk_p);
    scan_exclusive_single<<<1, 1, 0, stream>>>(blk_p, nbP, nI_p);
    scatter_partition<<<nbP, 256, 0, stream>>>(pInv, NPE, blk_p, nI_p, perm_p);

    auto repack = [&](const float* W, float* Wp, int Ktot, int N) {
        int total = (Ktot / 4) * N;
        repack_kernel<<<(total + 255) / 256, 256, 0, stream>>>(W, Wp, Ktot, N);
    };
    repack(entWIw, entWIp, 6 * 400, 160);
    repack(entWOw, entWOp, 6 * 400, 160);
    repack(relWIw, relWIp, 6 * 320, 160);
    repack(relWOw, relWOp, 6 * 320, 160);
    repack(entWSw, entWSp, 6 * 160, 160);
    repack(entWTw, entWTp, 6 * 80, 80);
    repack(relWSw, relWSp, 6 * 160, 160);
    repack(relWMw, relWMp, 6 * 160, 160);

    // ---- stage helpers ----
    auto ent_stage = [&](int li, const float* relIn, const float* entIn, const float* timIn,
                         float* entOut, float* timOut, int relu) {
        hipMemsetAsync(sum_e, 0, (size_t)N_ENT * 160 * 4, stream);
        edge_msg_kernel<<<NE / 16, 32, 0, stream>>>(
            relIn, 160, entIn, 160, timIn, 80,
            eRel, srcA, eTime, dstA, perm_e, nI_e,
            entWIp + (size_t)li * 400 * 160, entWOp + (size_t)li * 400 * 160,
            entWIb + li * 160, entWOb + li * 160, sum_e);
        dense_kernel<160><<<N_ENT / 16, 32, 0, stream>>>(
            entIn, entWSp + (size_t)li * 160 * 160, entWSb + li * 160,
            sum_e, recip_e, entOut, N_ENT, 160, relu);
        if (timOut)
            dense_kernel<80><<<(N_TIME + 15) / 16, 32, 0, stream>>>(
                timIn, entWTp + (size_t)li * 80 * 80, entWTb + li * 80,
                nullptr, nullptr, timOut, N_TIME, 80, relu);
    };
    auto rel_stage = [&](int li, const float* metaIn, const float* relIn,
                         float* relOut, float* metaOut, int relu) {
        hipMemsetAsync(sum_p, 0, (size_t)N_REL * 160 * 4, stream);
        edge_msg_kernel<<<NPE / 16, 32, 0, stream>>>(
            metaIn, 160, relIn, 160, nullptr, 0,
            pRel, pSrc, nullptr, pDst, perm_p, nI_p,
            relWIp + (size_t)li * 320 * 160, relWOp + (size_t)li * 320 * 160,
            relWIb + li * 160, relWOb + li * 160, sum_p);
        dense_kernel<160><<<N_REL / 16, 32, 0, stream>>>(
            relIn, relWSp + (size_t)li * 160 * 160, relWSb + li * 160,
            sum_p, recip_p, relOut, N_REL, 160, relu);
        if (metaOut)
            dense_kernel<160><<<N_META / 16, 32, 0, stream>>>(
                metaIn, relWMp + (size_t)li * 160 * 160, relWMb + li * 160,
                nullptr, nullptr, metaOut, N_META, 160, relu);
    };

    // ---- encoder ----
    ent_stage(0, rel_feat, ent_feat, time_emb, bufE_A, bufT, 1);
    rel_stage(0, meta_emb, rel_feat, bufR_A, bufM, 1);
    ent_stage(1, bufR_A, bufE_A, bufT, outEnt, outTim, 0);
    rel_stage(1, bufM, bufR_A, outRel, outMeta, 0);

    // ---- variational heads (idx 4 = mean, idx 5 = log_std) ----
    ent_stage(4, outRel, outEnt, outTim, bufE_A, nullptr, 0);  // mean_e
    rel_stage(4, outMeta, outRel, bufR_A, nullptr, 0);         // mean_r
    ent_stage(5, outRel, outEnt, outTim, bufE_B, nullptr, 0);  // log_std_e
    rel_stage(5, outMeta, outRel, bufR_B, nullptr, 0);         // log_std_r

    // ---- reparameterization (in place into mean buffers) ----
    sample_kernel<<<(N_ENT * 160 + 255) / 256, 256, 0, stream>>>(bufE_A, bufE_B, noise_e, N_ENT * 160);
    sample_kernel<<<(N_REL * 160 + 255) / 256, 256, 0, stream>>>(bufR_A, bufR_B, noise_r, N_REL * 160);

    // ---- decoder (weight idx 2, 3) ----
    ent_stage(2, bufR_A, bufE_A, outTim, bufE_B, bufT, 1);
    rel_stage(2, outMeta, bufR_A, bufR_B, bufM, 1);
    ent_stage(3, bufR_B, bufE_B, bufT, outEntO, outTimO, 0);
    rel_stage(3, bufM, bufR_B, outRelO, outMetaO, 0);
}